// VectorQuantizer_57664230916991
// MI455X (gfx1250) — compile-verified
//
#include <hip/hip_runtime.h>
#include <math.h>

#define B_ROWS 8192
#define DIMS   128
#define KCODES 32768
#define TOPK   8
#define ALPHA_ 0.25f
#define M_TILE 32   // z rows per block (two 16-row WMMA tiles)

typedef __attribute__((ext_vector_type(2))) float v2f;
typedef __attribute__((ext_vector_type(8))) float v8f;

// ---------------------------------------------------------------------------
// Prep: codebook row norms -> d_ws, zero the loss accumulator slot.
// ---------------------------------------------------------------------------
__global__ __launch_bounds__(256) void vq_prep(const float* __restrict__ cb,
                                               float* __restrict__ cnorm,
                                               float* __restrict__ loss_slot) {
  int t = blockIdx.x * 256 + threadIdx.x;
  if (blockIdx.x == 0 && threadIdx.x == 0) loss_slot[0] = 0.0f;
  if (t < KCODES) {
    const float4* row = (const float4*)(cb + (size_t)t * DIMS);
    float s = 0.f;
#pragma unroll
    for (int i = 0; i < DIMS / 4; ++i) {
      float4 v = row[i];
      s = fmaf(v.x, v.x, s); s = fmaf(v.y, v.y, s);
      s = fmaf(v.z, v.z, s); s = fmaf(v.w, v.w, s);
    }
    cnorm[t] = s;
  }
}

// branch-skipped max-replacement top-8 insertion
#define TOP8_INSERT(TD, TI, TMAX, TPOS, DV, N)                         \
  if ((DV) < (TMAX)) {                                                 \
    _Pragma("unroll")                                                  \
    for (int j = 0; j < TOPK; ++j)                                     \
      if (j == (TPOS)) { (TD)[j] = (DV); (TI)[j] = (N); }              \
    (TMAX) = (TD)[0]; (TPOS) = 0;                                      \
    _Pragma("unroll")                                                  \
    for (int j = 1; j < TOPK; ++j)                                     \
      if ((TD)[j] > (TMAX)) { (TMAX) = (TD)[j]; (TPOS) = j; }          \
  }

// ---------------------------------------------------------------------------
// Main fused kernel: one block = 32 z-rows (two A-tiles resident in VGPRs),
// 4 wave32s stripe the 2048 column tiles. Each B fragment feeds 2 WMMAs.
// ---------------------------------------------------------------------------
__global__ __launch_bounds__(128) void vq_main(const float* __restrict__ z,
                                               const float* __restrict__ cb,
                                               const float* __restrict__ cnorm,
                                               float* __restrict__ out) {
  __shared__ float tileb[4][M_TILE][17];          // per-wave 32x16 tile (padded)
  __shared__ float md[4 * 32 * 2 * TOPK];         // per-lane top-8 dists (2 rows)
  __shared__ int   mi[4 * 32 * 2 * TOPK];         // per-lane top-8 indices
  __shared__ float sw[M_TILE][TOPK];              // final softmax weights
  __shared__ int   si[M_TILE][TOPK];              // final indices

  const int tid  = threadIdx.x;
  const int w    = tid >> 5;        // wave id 0..3
  const int lane = tid & 31;
  const int rc   = lane & 15;       // A-row / B-col within tile
  const int hi   = lane >> 4;       // K half (WMMA f32 16x16x4 layout)
  const int m0   = blockIdx.x * M_TILE;

  // ---- Preload both A-tiles for all 32 k-steps (z rows stay in VGPRs) ----
  v2f afA[32], afB[32];
  {
    const float* zrA = z + (size_t)(m0 + rc) * DIMS + 2 * hi;
    const float* zrB = zrA + (size_t)16 * DIMS;
#pragma unroll
    for (int kk = 0; kk < 32; ++kk) {
      afA[kk].x = zrA[kk * 4 + 0];
      afA[kk].y = zrA[kk * 4 + 1];
      afB[kk].x = zrB[kk * 4 + 0];
      afB[kk].y = zrB[kk * 4 + 1];
    }
  }

  // ---- Private top-8 state for the two rows this lane scans ----
  float td0[TOPK], td1[TOPK]; int ti0[TOPK], ti1[TOPK];
#pragma unroll
  for (int j = 0; j < TOPK; ++j) {
    td0[j] = 3.0e38f; ti0[j] = 0;
    td1[j] = 3.0e38f; ti1[j] = 0;
  }
  float tmax0 = 3.0e38f, tmax1 = 3.0e38f;
  int   tpos0 = 0,        tpos1 = 0;

  for (int nt = w; nt < KCODES / 16; nt += 4) {
    const int n0 = nt * 16;
    v8f acc0 = {}, acc1 = {};
    const float* bb = cb + (size_t)(n0 + rc) * DIMS + 2 * hi;
#pragma unroll
    for (int kk = 0; kk < 32; ++kk) {
      v2f bf;
      bf.x = bb[kk * 4 + 0];
      bf.y = bb[kk * 4 + 1];
      acc0 = __builtin_amdgcn_wmma_f32_16x16x4_f32(
          false, afA[kk], false, bf, (short)0, acc0, false, false);
      acc1 = __builtin_amdgcn_wmma_f32_16x16x4_f32(
          false, afB[kk], false, bf, (short)0, acc1, false, false);
    }

    // Spill both 16x16 tiles to LDS so each lane can scan 8 columns of its two rows.
#pragma unroll
    for (int r = 0; r < 8; ++r) {
      tileb[w][r + 8 * hi][rc]      = acc0[r];
      tileb[w][16 + r + 8 * hi][rc] = acc1[r];
    }

    // Distance = ||c||^2 - 2 z.c   (row-constant ||z||^2 dropped: shift-inv.)
    const float* cn = cnorm + n0 + 8 * hi;
#pragma unroll
    for (int c = 0; c < 8; ++c) {
      float cnv = cn[c];
      int   n   = n0 + 8 * hi + c;
      float d0 = fmaf(-2.0f, tileb[w][rc][8 * hi + c], cnv);
      TOP8_INSERT(td0, ti0, tmax0, tpos0, d0, n);
      float d1 = fmaf(-2.0f, tileb[w][16 + rc][8 * hi + c], cnv);
      TOP8_INSERT(td1, ti1, tmax1, tpos1, d1, n);
    }
  }

  // ---- Publish per-lane lists (2 per lane), merge 8 lists per row ----
  {
    int base = (w * 32 + lane) * (2 * TOPK);
#pragma unroll
    for (int j = 0; j < TOPK; ++j) {
      md[base + j] = td0[j];        mi[base + j] = ti0[j];
      md[base + TOPK + j] = td1[j]; mi[base + TOPK + j] = ti1[j];
    }
  }
  __syncthreads();

  if (tid < M_TILE) {               // wave 0: one thread per output row
    const int r    = tid;
    const int half = r >> 4;        // which per-lane list (row / row+16)
    const int rr   = r & 15;
    float fd[TOPK]; int fi_[TOPK];
#pragma unroll
    for (int j = 0; j < TOPK; ++j) { fd[j] = 3.0e38f; fi_[j] = 0; }

    for (int w2 = 0; w2 < 4; ++w2)
      for (int h = 0; h < 2; ++h) {
        int src = (w2 * 32 + h * 16 + rr) * (2 * TOPK) + half * TOPK;
        for (int j = 0; j < TOPK; ++j) {
          float cd = md[src + j]; int ci = mi[src + j];
          if (cd < fd[TOPK - 1]) {
#pragma unroll
            for (int q = 0; q < TOPK; ++q) {  // sorted bubble insert
              if (cd < fd[q]) {
                float t1 = fd[q]; int t2 = fi_[q];
                fd[q] = cd; fi_[q] = ci; cd = t1; ci = t2;
              }
            }
          }
        }
      }

    // softmax over the 8 (stabilized by min distance; shift-invariant)
    float dmin = fd[0];
    float fw[TOPK]; float wsum = 0.f;
#pragma unroll
    for (int j = 0; j < TOPK; ++j) { fw[j] = __expf(dmin - fd[j]); wsum += fw[j]; }
    float inv = 1.0f / wsum;

    float* oidx = out + (size_t)B_ROWS * DIMS;
#pragma unroll
    for (int j = 0; j < TOPK; ++j) {
      fw[j] *= inv;
      sw[r][j] = fw[j];
      si[r][j] = fi_[j];
      oidx[(size_t)(m0 + r) * TOPK + j] = (float)fi_[j];
    }
  }
  __syncthreads();

  // ---- Gather z_q = sum_j w_j * C[idx_j]; write out; loss partials ----
  float lsum = 0.f;
#pragma unroll
  for (int halfrow = 0; halfrow < 2; ++halfrow) {
    const int gr  = (tid >> 3) + 16 * halfrow;   // row 0..31
    const int seg = (tid & 7) * 16;              // 16 dims per thread
    float accv[16];
#pragma unroll
    for (int d2 = 0; d2 < 16; ++d2) accv[d2] = 0.f;
#pragma unroll
    for (int j = 0; j < TOPK; ++j) {
      float wgt = sw[gr][j];
      const float4* crow = (const float4*)(cb + (size_t)si[gr][j] * DIMS + seg);
#pragma unroll
      for (int q = 0; q < 4; ++q) {
        float4 v = crow[q];
        accv[q * 4 + 0] = fmaf(wgt, v.x, accv[q * 4 + 0]);
        accv[q * 4 + 1] = fmaf(wgt, v.y, accv[q * 4 + 1]);
        accv[q * 4 + 2] = fmaf(wgt, v.z, accv[q * 4 + 2]);
        accv[q * 4 + 3] = fmaf(wgt, v.w, accv[q * 4 + 3]);
      }
    }
    const float* zr2 = z + (size_t)(m0 + gr) * DIMS + seg;
    float* outr = out + (size_t)(m0 + gr) * DIMS + seg;
#pragma unroll
    for (int d2 = 0; d2 < 16; ++d2) {
      float diff = accv[d2] - zr2[d2];
      lsum = fmaf(diff, diff, lsum);
      outr[d2] = accv[d2];
    }
  }
  // wave32 reduction then one atomic per wave
#pragma unroll
  for (int off = 16; off > 0; off >>= 1)
    lsum += __shfl_down(lsum, off, 32);
  if (lane == 0) {
    float* loss_slot = out + (size_t)B_ROWS * DIMS + (size_t)B_ROWS * TOPK;
    atomicAdd(loss_slot,
              lsum * ((1.0f + ALPHA_) / ((float)B_ROWS * (float)DIMS)));
  }
}

extern "C" void kernel_launch(void* const* d_in, const int* in_sizes, int n_in,
                              void* d_out, int out_size, void* d_ws, size_t ws_size,
                              hipStream_t stream) {
  const float* z  = (const float*)d_in[0];
  const float* cb = (const float*)d_in[1];
  float* out      = (float*)d_out;
  float* cnorm    = (float*)d_ws;   // KCODES floats (128 KB)
  float* loss_slot = out + (size_t)B_ROWS * DIMS + (size_t)B_ROWS * TOPK;

  vq_prep<<<KCODES / 256, 256, 0, stream>>>(cb, cnorm, loss_slot);
  vq_main<<<B_ROWS / M_TILE, 128, 0, stream>>>(z, cb, cnorm, out);
}